// ProCoMLoss_89206470738633
// MI455X (gfx1250) — compile-verified
//
#include <hip/hip_runtime.h>
#include <math.h>

#define N_BATCH     1024
#define FEAT        128
#define NUM_CLASSES 1000
#define MAX_MODES   10
#define NU          63            // FEAT/2 - 1
#define NCM         (NUM_CLASSES * MAX_MODES)   // 10000
#define COL_TILES   (NCM / 16)    // 625
#define ROW_TILES   (N_BATCH / 16) // 64
#define LDSA_STRIDE 132           // padded row stride (words) -> conflict-free ds_load_b64

typedef __attribute__((ext_vector_type(2))) float v2f;
typedef __attribute__((ext_vector_type(8))) float v8f;

// ---------- helpers ----------

__device__ __forceinline__ unsigned enc_f(float f) {
    unsigned u = __float_as_uint(f);
    return (u & 0x80000000u) ? ~u : (u | 0x80000000u);
}
__device__ __forceinline__ float dec_f(unsigned u) {
    return __uint_as_float((u & 0x80000000u) ? (u & 0x7fffffffu) : ~u);
}

// log(I0(x) * exp(-x)) via Abramowitz & Stegun 9.8.1 / 9.8.2
__device__ float log_i0e(float x) {
    if (x < 3.75f) {
        float t = x * (1.0f / 3.75f); t *= t;
        float i0 = 1.0f + t * (3.5156229f + t * (3.0899424f + t * (1.2067492f +
                   t * (0.2659732f + t * (0.0360768f + t * 0.0045813f)))));
        return __logf(i0) - x;
    } else {
        float t = 3.75f / x;
        float p = 0.39894228f + t * (0.01328592f + t * (0.00225319f + t * (-0.00157565f +
                  t * (0.00916281f + t * (-0.02057706f + t * (0.02635537f +
                  t * (-0.01647633f + t * 0.00392377f)))))));
        return __logf(p) - 0.5f * __logf(x);
    }
}

#define RESCALE_THR  0x1p60f
#define RESCALE_MUL  0x1p-60f
#define LN2_60       41.588830833596718f   // 60 * ln(2)

// Linear-domain Miller downward recurrence with power-of-two rescaling.
// Equivalent to the reference's log-domain scan; returns log Ive_NU(x).
// Scalar version (used per-prototype in k_proto).
__device__ float log_ive_nu(float x) {
    float invx = 1.0f / x;
    float I = 1.0f, I1 = 0.0f, cnt = 0.0f;
    float fi2 = 2.0f * (float)(2 * NU);
    #pragma unroll 1
    for (int i = 2 * NU; i > NU; --i) {          // i = 126 .. 64
        float nI = fmaf(fi2 * invx, I, I1);
        fi2 -= 2.0f;
        I1 = I; I = nI;
        if (I > RESCALE_THR) { I *= RESCALE_MUL; I1 *= RESCALE_MUL; cnt += 1.0f; }
    }
    float sI = I, sc = cnt;                      // I~_63 (scaled)
    #pragma unroll 1
    for (int i = NU; i >= 1; --i) {              // i = 63 .. 1
        float nI = fmaf(fi2 * invx, I, I1);
        fi2 -= 2.0f;
        I1 = I; I = nI;
        if (I > RESCALE_THR) { I *= RESCALE_MUL; I1 *= RESCALE_MUL; cnt += 1.0f; }
    }
    return log_i0e(x) + __logf(sI) - __logf(I) + (sc - cnt) * LN2_60;
}

// ---------- kernel 1: init scratch + encoded output ----------

__global__ void k_init(float* sum_oh, float* wfeat, unsigned* out_enc) {
    int i = blockIdx.x * blockDim.x + threadIdx.x;
    if (i < NCM) sum_oh[i] = 0.0f;
    if (i < NCM * FEAT) wfeat[i] = 0.0f;
    if (i < N_BATCH * NUM_CLASSES) out_enc[i] = 0x007FFFFFu;   // enc(-inf)
}

// ---------- kernel 2: soft assignment (softmax over modes) ----------

__global__ void k_soft(const float* __restrict__ F, const float* __restrict__ Ave_old,
                       const int* __restrict__ labels, float* __restrict__ soft) {
    __shared__ float red[128];
    __shared__ float num[MAX_MODES], den[MAX_MODES];
    __shared__ float nf;
    int n = blockIdx.x, t = threadIdx.x;
    int c = labels[n];
    float f = F[n * FEAT + t];

    red[t] = f * f; __syncthreads();
    for (int s = 64; s > 0; s >>= 1) { if (t < s) red[t] += red[t + s]; __syncthreads(); }
    if (t == 0) nf = sqrtf(red[0]);
    __syncthreads();

    for (int m = 0; m < MAX_MODES; ++m) {
        float a = Ave_old[((size_t)c * MAX_MODES + m) * FEAT + t];
        red[t] = f * a; __syncthreads();
        for (int s = 64; s > 0; s >>= 1) { if (t < s) red[t] += red[t + s]; __syncthreads(); }
        if (t == 0) num[m] = red[0];
        __syncthreads();
        red[t] = a * a; __syncthreads();
        for (int s = 64; s > 0; s >>= 1) { if (t < s) red[t] += red[t + s]; __syncthreads(); }
        if (t == 0) den[m] = fmaxf(nf * sqrtf(red[0]), 1e-8f);
        __syncthreads();
    }

    if (t == 0) {
        float l[MAX_MODES], mx = -INFINITY, ss = 0.0f;
        for (int m = 0; m < MAX_MODES; ++m) { l[m] = num[m] / den[m]; mx = fmaxf(mx, l[m]); }
        for (int m = 0; m < MAX_MODES; ++m) { l[m] = __expf(l[m] - mx); ss += l[m]; }
        for (int m = 0; m < MAX_MODES; ++m) soft[n * MAX_MODES + m] = l[m] / ss;
    }
}

// ---------- kernel 3: segment-sum scatter ----------

__global__ void k_scatter(const float* __restrict__ F, const float* __restrict__ soft,
                          const int* __restrict__ labels,
                          float* sum_oh, float* wfeat) {
    int n = blockIdx.x, t = threadIdx.x;
    int c = labels[n];
    float f = F[n * FEAT + t];
    for (int m = 0; m < MAX_MODES; ++m) {
        float s = soft[n * MAX_MODES + m];
        atomicAdd(&wfeat[((size_t)c * MAX_MODES + m) * FEAT + t], s * f);
    }
    if (t < MAX_MODES) atomicAdd(&sum_oh[c * MAX_MODES + t], soft[n * MAX_MODES + t]);
}

// ---------- kernel 4: prototype blend, kappa, logc, normalized prototypes ----------

__global__ void k_proto(const float* __restrict__ Ave_old, const float* __restrict__ Amount_old,
                        const float* __restrict__ sum_oh, const float* __restrict__ wfeat,
                        float* __restrict__ aven, float* __restrict__ kappa,
                        float* __restrict__ logc) {
    __shared__ float red[128];
    __shared__ float Rsh;
    int j = blockIdx.x, t = threadIdx.x;

    float s = sum_oh[j];
    float amt = Amount_old[j];
    float w = s / (s + amt);
    if (!(w == w)) w = 0.0f;
    float cma = wfeat[(size_t)j * FEAT + t] / ((s == 0.0f) ? 1.0f : s);
    float av  = Ave_old[(size_t)j * FEAT + t] * (1.0f - w) + cma * w;

    red[t] = av * av; __syncthreads();
    for (int st = 64; st > 0; st >>= 1) { if (t < st) red[t] += red[t + st]; __syncthreads(); }
    if (t == 0) Rsh = sqrtf(red[0]);
    __syncthreads();
    float R = Rsh;

    aven[(size_t)j * FEAT + t] = av / fmaxf(R, 1e-12f);

    if (t == 0) {
        float kp = (float)FEAT * R / (1.0f - R * R);
        if (kp > 1e5f || kp < 0.0f || !(kp == kp)) kp = 1e5f;
        kappa[j] = kp;
        float lognu = log_ive_nu(kp);
        logc[j] = lognu + kp - (float)NU * __logf(kp + 1e-20f);
    }
}

// ---------- kernel 5: WMMA f32 GEMM + vMF log-likelihood epilogue + mode-max ----------
// Grid: (79 col-blocks of 8 tiles, 64 row-blocks). All 8 waves of a block share the
// same 16-row A tile, staged once in LDS (padded stride -> conflict-free ds_load_b64).
// K=128 via 32x v_wmma_f32_16x16x4_f32 per tile. Epilogue runs 8 independent
// Bessel recurrences per lane in lock-step for ILP; only 4 v_log_f32 per element.

__global__ void __launch_bounds__(256)
k_wmma(const float* __restrict__ F, const float* __restrict__ aven,
       const float* __restrict__ kappa, const float* __restrict__ logc,
       unsigned* __restrict__ out_enc) {
    __shared__ float ldsA[16 * LDSA_STRIDE];

    const int t  = threadIdx.x;
    const int m0 = blockIdx.y << 4;

    // ---- stage A tile (16 x 128 f32) into LDS, float4 per transfer ----
    {
        const float4* Fv = (const float4*)(F + (size_t)m0 * FEAT);
        #pragma unroll
        for (int j = 0; j < 2; ++j) {
            int idx = t + (j << 8);          // 0..511 float4 slots
            int row = idx >> 5;              // 32 float4 per row
            int q   = idx & 31;
            float4 d = Fv[row * 32 + q];
            *(float4*)&ldsA[row * LDSA_STRIDE + (q << 2)] = d;
        }
    }
    __syncthreads();

    const int wib = t >> 5;                  // wave in block: 0..7
    const int tn  = blockIdx.x * 8 + wib;    // column tile
    if (tn >= COL_TILES) return;             // whole wave exits together

    const int lane = t & 31;
    const int hi   = lane >> 4;              // 0: K {0,1}, 1: K {2,3}
    const int l16  = lane & 15;
    const int n0   = tn << 4;
    const int col  = n0 + l16;

    const float* ar = ldsA + l16 * LDSA_STRIDE;
    const float* br = aven + (size_t)col * FEAT;

    v8f acc = {};
    #pragma unroll 8
    for (int kb = 0; kb < 32; ++kb) {
        int k0 = (kb << 2) + (hi << 1);
        v2f a; a.x = ar[k0]; a.y = ar[k0 + 1];
        v2f b; b.x = br[k0]; b.y = br[k0 + 1];
        acc = __builtin_amdgcn_wmma_f32_16x16x4_f32(false, a, false, b,
                                                    (short)0, acc, false, false);
    }

    // ---- epilogue: 8 elements per lane, lock-step recurrences for ILP ----
    float kf = kappa[col];
    float lc = logc[col];
    float kf2 = kf * kf;
    int cc = col / MAX_MODES;

    float kn[8], invx[8], I[8], I1[8], cnt[8];
    #pragma unroll
    for (int v = 0; v < 8; ++v) {
        float sim = acc[v];                                   // TEMPERATURE == 1.0
        kn[v]   = sqrtf(fmaxf(kf2 + 2.0f * kf * sim + 1.0f, 0.0f));
        invx[v] = 1.0f / kn[v];
        I[v] = 1.0f; I1[v] = 0.0f; cnt[v] = 0.0f;
    }

    float fi2 = 2.0f * (float)(2 * NU);
    #pragma unroll 1
    for (int i = 2 * NU; i > NU; --i) {       // 126 .. 64
        #pragma unroll
        for (int v = 0; v < 8; ++v) {
            float nI = fmaf(fi2 * invx[v], I[v], I1[v]);
            I1[v] = I[v]; I[v] = nI;
            if (I[v] > RESCALE_THR) { I[v] *= RESCALE_MUL; I1[v] *= RESCALE_MUL; cnt[v] += 1.0f; }
        }
        fi2 -= 2.0f;
    }
    float sI[8], sc[8];
    #pragma unroll
    for (int v = 0; v < 8; ++v) { sI[v] = I[v]; sc[v] = cnt[v]; }   // I~_63
    #pragma unroll 1
    for (int i = NU; i >= 1; --i) {           // 63 .. 1
        #pragma unroll
        for (int v = 0; v < 8; ++v) {
            float nI = fmaf(fi2 * invx[v], I[v], I1[v]);
            I1[v] = I[v]; I[v] = nI;
            if (I[v] > RESCALE_THR) { I[v] *= RESCALE_MUL; I1[v] *= RESCALE_MUL; cnt[v] += 1.0f; }
        }
        fi2 -= 2.0f;
    }

    #pragma unroll 1
    for (int v = 0; v < 8; ++v) {
        float lognu = log_i0e(kn[v]) + __logf(sI[v]) - __logf(I[v])
                    + (sc[v] - cnt[v]) * LN2_60;
        float logit = lognu + kn[v] - (float)NU * __logf(kn[v] + 1e-20f) - lc;
        int r = m0 + v + (hi << 3);
        atomicMax(&out_enc[(size_t)r * NUM_CLASSES + cc], enc_f(logit));
    }
}

// ---------- kernel 6: decode encoded-max back to float ----------

__global__ void k_decode(unsigned* __restrict__ buf) {
    int i = blockIdx.x * blockDim.x + threadIdx.x;
    if (i < N_BATCH * NUM_CLASSES) {
        float f = dec_f(buf[i]);
        ((float*)buf)[i] = f;
    }
}

// ---------- launch ----------

extern "C" void kernel_launch(void* const* d_in, const int* in_sizes, int n_in,
                              void* d_out, int out_size, void* d_ws, size_t ws_size,
                              hipStream_t stream) {
    const float* F          = (const float*)d_in[0];
    const float* Ave_old    = (const float*)d_in[1];
    const float* Amount_old = (const float*)d_in[2];
    const int*   labels     = (const int*)d_in[3];

    float* ws     = (float*)d_ws;
    float* soft   = ws;                    // 10240
    float* sum_oh = soft   + 10240;        // 10000
    float* kappa  = sum_oh + 10000;        // 10000
    float* logc   = kappa  + 10000;        // 10000
    float* wfeat  = logc   + 10000;        // 1,280,000
    float* aven   = wfeat  + (size_t)NCM * FEAT;   // 1,280,000
    unsigned* out_enc = (unsigned*)d_out;

    // 1) init scratch + encoded -inf output (covers max extent 1,280,000)
    k_init<<<(NCM * FEAT + 255) / 256, 256, 0, stream>>>(sum_oh, wfeat, out_enc);
    // 2) soft assignments
    k_soft<<<N_BATCH, 128, 0, stream>>>(F, Ave_old, labels, soft);
    // 3) label scatter
    k_scatter<<<N_BATCH, 128, 0, stream>>>(F, soft, labels, sum_oh, wfeat);
    // 4) prototype update + kappa/logc + normalized prototypes (GEMM B)
    k_proto<<<NCM, 128, 0, stream>>>(Ave_old, Amount_old, sum_oh, wfeat, aven, kappa, logc);
    // 5) WMMA sims + vMF epilogue + per-class max
    {
        dim3 grid((COL_TILES + 7) / 8, ROW_TILES);
        k_wmma<<<grid, 256, 0, stream>>>(F, aven, kappa, logc, out_enc);
    }
    // 6) decode
    k_decode<<<(N_BATCH * NUM_CLASSES + 255) / 256, 256, 0, stream>>>(out_enc);
}